// GlowTts_38345468019234
// MI455X (gfx1250) — compile-verified
//
#include <hip/hip_runtime.h>
#include <hip/hip_bf16.h>
#include <cstdint>
#include <cstddef>

#define Bb 32
#define Cc 80
#define TXx 512
#define TYy 2048
#define K2C (2 * Cc)          // 160
#define NEGV (-1e9f)
#define LOG_2PI 1.8378770664093453f

typedef __attribute__((ext_vector_type(2))) float v2f;
typedef __attribute__((ext_vector_type(8))) float v8f;

// ---------------- workspace layout (bytes, 256-aligned) ----------------
static constexpr size_t OFF_AMAT  = 0;                                            // [B,TX,160] f32  (10.0 MB)
static constexpr size_t OFF_LP14  = OFF_AMAT  + (size_t)Bb * TXx * K2C * 4;       // [B,TX] f32
static constexpr size_t OFF_ZT    = OFF_LP14  + (size_t)Bb * TXx * 4;             // [B,TY,160] f32 (40 MB)
static constexpr size_t OFF_LOGPT = OFF_ZT    + (size_t)Bb * TYy * K2C * 4;       // [B,TY,TX] f32  (128 MB)
static constexpr size_t OFF_DIRB  = OFF_LOGPT + (size_t)Bb * TYy * TXx * 4;       // [B,TY,TX/32] u32 (4 MB)
static constexpr size_t OFF_XIDX  = OFF_DIRB  + (size_t)Bb * TYy * (TXx / 32) * 4;// [B,TY] i32
static constexpr size_t OFF_CNT   = OFF_XIDX  + (size_t)Bb * TYy * 4;             // [B,TX] i32

// ---------------- kernel 0a: A-matrix (scale | mean*scale) + lp14 ----------------
__global__ __launch_bounds__(256) void k_prep_a(const float* __restrict__ o_mean,
                                                const float* __restrict__ o_log_scale,
                                                float* __restrict__ Amat,
                                                float* __restrict__ lp14) {
    int i = blockIdx.x * blockDim.x + threadIdx.x;   // i = b*TX + x
    if (i >= Bb * TXx) return;
    int b = i / TXx;
    int x = i - b * TXx;
    const float* ols = o_log_scale + (size_t)b * Cc * TXx + x;
    const float* om  = o_mean      + (size_t)b * Cc * TXx + x;
    float* arow = Amat + (size_t)i * K2C;
    float acc = 0.f;
#pragma unroll 4
    for (int c = 0; c < Cc; ++c) {
        float l = ols[(size_t)c * TXx];
        float m = om[(size_t)c * TXx];
        float s = __expf(-2.f * l);
        float ms = m * s;
        arow[c]      = s;
        arow[Cc + c] = ms;
        acc += -0.5f * LOG_2PI - l - 0.5f * m * ms;   // logp1 + logp4 terms
    }
    lp14[i] = acc;
}

// ---------------- kernel 0b: packed B operand ZT[B,TY,160] + masked z output ----------------
// ZT[b][y][k] = (k < C) ? -0.5*zm^2 : zm   with zm = z[b,c,y]*y_mask, c = k mod C.
__global__ __launch_bounds__(256) void k_prep_z(const float* __restrict__ z,
                                                const int* __restrict__ ylens,
                                                float* __restrict__ ZT,
                                                float* __restrict__ zout) {
    int i = blockIdx.x * blockDim.x + threadIdx.x;   // i = b*TY + y
    if (i >= Bb * TYy) return;
    int b = i / TYy;
    int y = i - b * TYy;
    bool ym = y < ylens[b];
    const float* zb = z + (size_t)b * Cc * TYy + y;
    float* zo = zout + (size_t)b * Cc * TYy + y;
    float* zr = ZT + (size_t)i * K2C;
#pragma unroll 4
    for (int c = 0; c < Cc; ++c) {
        float zv = ym ? zb[(size_t)c * TYy] : 0.f;
        zo[(size_t)c * TYy] = zv;
        zr[c]      = -0.5f * zv * zv;
        zr[Cc + c] = zv;
    }
}

// ---------------- kernel 1: logp via V_WMMA_F32_16X16X4_F32 ----------------
// grid: (TY/64, TX/16, B); one wave32 per 16x64 strip (4 accumulators, A frag reused 4x).
// A frag (16x4 f32): lane m=lane&15 is row M; VGPR pair holds K = kb+2*half, kb+2*half+1.
// B frag (4x16 f32): lane n=lane&15 is col N; VGPR pair holds K rows kb+2*half, kb+2*half+1.
// C/D (16x16 f32): acc[j] = element (M = j + 8*half, N = lane&15).
__global__ __launch_bounds__(32) void k_logp_wmma(const float* __restrict__ Amat,
                                                  const float* __restrict__ ZT,
                                                  const float* __restrict__ lp14,
                                                  const int* __restrict__ xlens,
                                                  const int* __restrict__ ylens,
                                                  float* __restrict__ logpT) {
    const int b    = blockIdx.z;
    const int x0   = blockIdx.y * 16;
    const int y0   = blockIdx.x * 64;
    const int lane = threadIdx.x;
    const int half = lane >> 4;
    const int mn   = lane & 15;
    const int xlen = xlens[b];
    const int ylen = ylens[b];

    const float* Arow = Amat + ((size_t)(b * TXx + x0 + mn)) * K2C + 2 * half;
    const float* Zrow = ZT   + ((size_t)(b * TYy + y0 + mn)) * K2C + 2 * half;

    v8f accs[4] = {{}, {}, {}, {}};
#pragma unroll 2
    for (int kb = 0; kb < K2C; kb += 4) {
        v2f a  = *(const v2f*)(Arow + kb);                 // b64 load, reused 4x
        v2f b0 = *(const v2f*)(Zrow + kb);                 // b64 loads (packed K-pairs)
        v2f b1 = *(const v2f*)(Zrow + 16 * K2C + kb);
        v2f b2 = *(const v2f*)(Zrow + 32 * K2C + kb);
        v2f b3 = *(const v2f*)(Zrow + 48 * K2C + kb);
        accs[0] = __builtin_amdgcn_wmma_f32_16x16x4_f32(false, a, false, b0, (short)0, accs[0], false, false);
        accs[1] = __builtin_amdgcn_wmma_f32_16x16x4_f32(false, a, false, b1, (short)0, accs[1], false, false);
        accs[2] = __builtin_amdgcn_wmma_f32_16x16x4_f32(false, a, false, b2, (short)0, accs[2], false, false);
        accs[3] = __builtin_amdgcn_wmma_f32_16x16x4_f32(false, a, false, b3, (short)0, accs[3], false, false);
    }

    // epilogue: unconditional lp14 vector loads, branch-free masked stores
    const int rb = x0 + 8 * half;                          // first of this lane's 8 rows
    float4 lp0 = *(const float4*)(lp14 + b * TXx + rb);
    float4 lp1 = *(const float4*)(lp14 + b * TXx + rb + 4);
    float lp[8] = {lp0.x, lp0.y, lp0.z, lp0.w, lp1.x, lp1.y, lp1.z, lp1.w};
    bool xm[8];
#pragma unroll
    for (int j = 0; j < 8; ++j) xm[j] = (rb + j) < xlen;

#pragma unroll
    for (int t = 0; t < 4; ++t) {
        const int y = y0 + 16 * t + mn;
        const bool yok = y < ylen;
        float vals[8];
#pragma unroll
        for (int j = 0; j < 8; ++j) {
            float v = accs[t][j] + lp[j];
            vals[j] = (yok && xm[j]) ? v : 0.f;            // value * attn_mask
        }
        float* dst = logpT + ((size_t)b * TYy + y) * TXx + rb;  // transposed store
        *(float4*)(dst)     = make_float4(vals[0], vals[1], vals[2], vals[3]);
        *(float4*)(dst + 4) = make_float4(vals[4], vals[5], vals[6], vals[7]);
    }
}

// ---------------- kernel 2: Viterbi forward scan (sequential over TY) ----------------
__global__ __launch_bounds__(TXx) void k_viterbi_fwd(const float* __restrict__ logpT,
                                                     const int* __restrict__ xlens,
                                                     const int* __restrict__ ylens,
                                                     unsigned* __restrict__ dirbits) {
    __shared__ float v[2][TXx];                           // ping-pong: 1 barrier/step
    const int b = blockIdx.x;
    const int x = threadIdx.x;
    const int xlen = xlens[b];
    const int ylen = ylens[b];
    v[0][x] = 0.f;
    __syncthreads();
    const float* colbase = logpT + (size_t)b * TYy * TXx;
    unsigned* dbase = dirbits + (size_t)b * TYy * (TXx / 32);
    int p = 0;
    for (int j = 0; j < TYy; ++j) {
        float col = colbase[(size_t)j * TXx + x];
        if (j + 1 < TYy) __builtin_prefetch(colbase + (size_t)(j + 1) * TXx + x, 0, 0);
        float vc = v[p][x];
        float v0 = (x == 0) ? NEGV : v[p][x - 1];
        bool dir  = vc >= v0;                             // true = stay at same x
        float vmx = dir ? vc : v0;
        float vnw = (x <= j) ? (vmx + col) : NEGV;
        v[p ^ 1][x] = vnw;
        bool dirm = (x >= xlen || j >= ylen) ? true : dir;  // force 'stay' outside mask
        unsigned m32 = (unsigned)__ballot(dirm ? 1 : 0);    // wave32: 1 bit/lane
        if ((x & 31) == 0) dbase[(size_t)j * (TXx / 32) + (x >> 5)] = m32;
        __syncthreads();
        p ^= 1;
    }
}

// ---------------- kernel 3: backtrack (1 lane per batch) ----------------
__global__ __launch_bounds__(32) void k_backtrack(const unsigned* __restrict__ dirbits,
                                                  const int* __restrict__ xlens,
                                                  const int* __restrict__ ylens,
                                                  int* __restrict__ xidx,
                                                  int* __restrict__ cnt) {
    const int b = threadIdx.x;
    if (b >= Bb) return;
    const int xlen = xlens[b];
    const int ylen = ylens[b];
    int* cb = cnt + b * TXx;
    for (int x = 0; x < TXx; ++x) cb[x] = 0;
    const unsigned* db = dirbits + (size_t)b * TYy * (TXx / 32);
    int idx = xlen - 1;
    for (int j = TYy - 1; j >= 0; --j) {
        xidx[b * TYy + j] = idx;
        if (j < ylen && idx >= 0 && idx < xlen) cb[idx] += 1;
        int ic = idx < 0 ? 0 : (idx > TXx - 1 ? TXx - 1 : idx);
        unsigned w = db[(size_t)j * (TXx / 32) + (ic >> 5)];
        int step = (int)((w >> (ic & 31)) & 1u);
        idx += step - 1;                                  // stay (1) or move left (0)
    }
}

// ---------------- kernel 4a: y_mean / y_log_scale gathers ----------------
__global__ __launch_bounds__(256) void k_gather_y(const float* __restrict__ o_mean,
                                                  const float* __restrict__ o_log_scale,
                                                  const int* __restrict__ xidx,
                                                  const int* __restrict__ xlens,
                                                  const int* __restrict__ ylens,
                                                  float* __restrict__ y_mean,
                                                  float* __restrict__ y_ls) {
    size_t i = (size_t)blockIdx.x * blockDim.x + threadIdx.x;
    if (i >= (size_t)Bb * Cc * TYy) return;
    int y  = (int)(i % TYy);
    int bc = (int)(i / TYy);
    int b  = bc / Cc;
    int xi = xidx[b * TYy + y];
    bool ok = (y < ylens[b]) && (xi >= 0) && (xi < xlens[b]);
    int xc = xi < 0 ? 0 : (xi > TXx - 1 ? TXx - 1 : xi);
    size_t src = (size_t)bc * TXx + xc;
    y_mean[i] = ok ? o_mean[src] : 0.f;
    y_ls[i]   = ok ? o_log_scale[src] : 0.f;
}

// ---------------- kernel 4b: attn_out one-hot [B,TY,TX] ----------------
__global__ __launch_bounds__(256) void k_attn(const int* __restrict__ xidx,
                                              const int* __restrict__ xlens,
                                              const int* __restrict__ ylens,
                                              float* __restrict__ attn) {
    size_t i = (size_t)blockIdx.x * blockDim.x + threadIdx.x;
    if (i >= (size_t)Bb * TYy * TXx) return;
    int x  = (int)(i % TXx);
    int by = (int)(i / TXx);                  // by = b*TY + y
    int b  = by / TYy;
    int y  = by - b * TYy;
    int xi = xidx[by];
    attn[i] = (x == xi && y < ylens[b] && x < xlens[b]) ? 1.f : 0.f;
}

// ---------------- kernel 4c: durations ----------------
__global__ __launch_bounds__(256) void k_dur(const float* __restrict__ o_dur_log,
                                             const int* __restrict__ cnt,
                                             const int* __restrict__ xlens,
                                             float* __restrict__ out_dur_log,
                                             float* __restrict__ out_attn_dur) {
    int i = blockIdx.x * blockDim.x + threadIdx.x;
    if (i >= Bb * TXx) return;
    int b = i / TXx;
    int x = i - b * TXx;
    out_dur_log[i]  = o_dur_log[i];
    out_attn_dur[i] = (x < xlens[b]) ? log1pf((float)cnt[i]) : 0.f;
}

// ---------------- launch ----------------
extern "C" void kernel_launch(void* const* d_in, const int* in_sizes, int n_in,
                              void* d_out, int out_size, void* d_ws, size_t ws_size,
                              hipStream_t stream) {
    const float* o_mean      = (const float*)d_in[0];
    const float* o_log_scale = (const float*)d_in[1];
    const float* o_dur_log   = (const float*)d_in[2];
    const float* z           = (const float*)d_in[3];
    const int*   xlens       = (const int*)d_in[4];
    const int*   ylens       = (const int*)d_in[5];

    float* out = (float*)d_out;                       // concatenated in return order
    float* out_z       = out;
    float* out_ymean   = out_z     + (size_t)Bb * Cc * TYy;
    float* out_yls     = out_ymean + (size_t)Bb * Cc * TYy;
    float* out_attn    = out_yls   + (size_t)Bb * Cc * TYy;
    float* out_durlog  = out_attn  + (size_t)Bb * TYy * TXx;
    float* out_attndur = out_durlog + (size_t)Bb * TXx;

    char* ws = (char*)d_ws;
    float*    Amat    = (float*)(ws + OFF_AMAT);
    float*    lp14    = (float*)(ws + OFF_LP14);
    float*    ZT      = (float*)(ws + OFF_ZT);
    float*    logpT   = (float*)(ws + OFF_LOGPT);
    unsigned* dirbits = (unsigned*)(ws + OFF_DIRB);
    int*      xidx    = (int*)(ws + OFF_XIDX);
    int*      cnt     = (int*)(ws + OFF_CNT);

    // 0) operand prep (A packed per-x, B packed per-y, masked z to output)
    k_prep_a<<<(Bb * TXx + 255) / 256, 256, 0, stream>>>(o_mean, o_log_scale, Amat, lp14);
    k_prep_z<<<(Bb * TYy + 255) / 256, 256, 0, stream>>>(z, ylens, ZT, out_z);

    // 1) logp GEMM (WMMA f32 16x16x4), 16x64 strip per wave, stored transposed + masked
    dim3 g1(TYy / 64, TXx / 16, Bb);
    k_logp_wmma<<<g1, 32, 0, stream>>>(Amat, ZT, lp14, xlens, ylens, logpT);

    // 2) Viterbi forward (sequential over TY), directions as bitmasks
    k_viterbi_fwd<<<Bb, TXx, 0, stream>>>(logpT, xlens, ylens, dirbits);

    // 3) backtrack -> path index per y + per-x counts
    k_backtrack<<<1, 32, 0, stream>>>(dirbits, xlens, ylens, xidx, cnt);

    // 4) outputs: gathers, one-hot attn, durations
    k_gather_y<<<(int)(((size_t)Bb * Cc * TYy + 255) / 256), 256, 0, stream>>>(
        o_mean, o_log_scale, xidx, xlens, ylens, out_ymean, out_yls);
    k_attn<<<(int)(((size_t)Bb * TYy * TXx + 255) / 256), 256, 0, stream>>>(
        xidx, xlens, ylens, out_attn);
    k_dur<<<(Bb * TXx + 255) / 256, 256, 0, stream>>>(
        o_dur_log, cnt, xlens, out_durlog, out_attndur);
}